// HyperbolicPoincareDynamics_33998961115175
// MI455X (gfx1250) — compile-verified
//
#include <hip/hip_runtime.h>
#include <math.h>

// ---------------------------------------------------------------------------
// HyperbolicPoincareDynamics fused kernel for gfx1250 (MI455X).
//  - f32 WMMA (v_wmma_f32_16x16x4_f32) for u @ z  (34.4 GFLOP fp32)
//  - Tensor Data Mover (tensor_load_to_lds) stages z panels into LDS
//  - whole pipeline fused: HBM traffic = 2 reads + 2 writes of [B,512] only
// ---------------------------------------------------------------------------

typedef __attribute__((ext_vector_type(2))) float        v2f;
typedef __attribute__((ext_vector_type(8))) float        v8f;
typedef __attribute__((ext_vector_type(4))) unsigned int v4u;
typedef __attribute__((ext_vector_type(8))) int          v8i;
typedef __attribute__((ext_vector_type(4))) int          v4i;

#define BN      65536
#define DD      512
#define FEPS    1e-15f
#define PMAXN   (1.0f - 4e-3f)

#define ROWS    32            // rows per workgroup
#define PANEL   64            // z columns staged per TDM panel
#define USTR    516           // padded LDS row stride (floats), 516%64==4 -> conflict-free A reads

// dynamic LDS layout (bytes)
#define ZS_OFF  0
#define ZS_B    (DD * PANEL * 4)        // 131072: raw z panel [k][n]
#define US_OFF  (ZS_OFF + ZS_B)
#define US_B    (ROWS * USTR * 4)       // 66048 : u tile  (WMMA A operand)
#define WS_OFF  (US_OFF + US_B)
#define WS_B    (ROWS * USTR * 4)       // 66048 : w = sinh(v) tile
#define CX_OFF  (WS_OFF + WS_B)
#define SMEM_B  (CX_OFF + ROWS * 4)     // + cx2[32]  => 263296 bytes total

// ---------------------------- small device helpers -------------------------
__device__ __forceinline__ float wred(float v) {
#pragma unroll
  for (int o = 16; o > 0; o >>= 1) v += __shfl_xor(v, o, 32);
  return v;
}
__device__ __forceinline__ float sigmoidf_(float x) { return 1.0f / (1.0f + expf(-x)); }
__device__ __forceinline__ float atanhc_(float x) {
  x = fminf(fmaxf(x, -1.0f + 1e-7f), 1.0f - 1e-7f);
  return atanhf(x);
}
__device__ __forceinline__ float d4(const float4& a, const float4& b) {
  return a.x * b.x + a.y * b.y + a.z * b.z + a.w * b.w;
}
__device__ __forceinline__ float4 axpby4(float s, const float4& a, float t, const float4& b) {
  float4 r; r.x = s * a.x + t * b.x; r.y = s * a.y + t * b.y;
  r.z = s * a.z + t * b.z; r.w = s * a.w + t * b.w; return r;
}
__device__ __forceinline__ float4 ax4(float s, const float4& a) {
  float4 r; r.x = s * a.x; r.y = s * a.y; r.z = s * a.z; r.w = s * a.w; return r;
}

// ---------------------------- prep: column stats of z ----------------------
__global__ void hpd_prep(const float* __restrict__ z, const float* __restrict__ bias,
                         float* __restrict__ zn, float* __restrict__ rzn,
                         float* __restrict__ c2r, float* __restrict__ s2r) {
  int o = blockIdx.x * blockDim.x + threadIdx.x;
  if (o >= DD) return;
  float s = 0.0f;
#pragma unroll 4
  for (int k = 0; k < DD; ++k) { float v = z[(size_t)k * DD + o]; s += v * v; }
  float n = sqrtf(fmaxf(s, FEPS));
  zn[o] = n; rzn[o] = 1.0f / n;
  float tb = 2.0f * bias[o];
  c2r[o] = coshf(tb); s2r[o] = sinhf(tb);
}

// ---------------------------- main fused kernel ----------------------------
__global__ __launch_bounds__(256, 1)
void hpd_main(const float* __restrict__ xcur, const float* __restrict__ xprev,
              const float* __restrict__ z,
              const float* __restrict__ zn, const float* __restrict__ rzn,
              const float* __restrict__ c2r, const float* __restrict__ s2r,
              const float* __restrict__ alpha_p, const float* __restrict__ step_p,
              float* __restrict__ out) {
  extern __shared__ char smem[];
  float* zS  = (float*)(smem + ZS_OFF);   // [512][64] raw z panel
  float* uS  = (float*)(smem + US_OFF);   // [32][USTR]
  float* wS  = (float*)(smem + WS_OFF);   // [32][USTR]
  float* cxS = (float*)(smem + CX_OFF);   // [32]

  const int tid  = threadIdx.x;
  const int wave = tid >> 5;
  const int lane = tid & 31;
  const int row0 = blockIdx.x * ROWS;

  // ===== Phase 1: u = expmap0(logmap(x_prev, x_cur)) per row ===============
  // wave w handles rows 4w..4w+3; lane handles float4 chunks lane,+32,+64,+96
  for (int rr = 0; rr < 4; ++rr) {
    const int r = wave * 4 + rr;
    const float4* xc4 = (const float4*)(xcur + (size_t)(row0 + r) * DD);
    const float4* xp4 = (const float4*)(xprev + (size_t)(row0 + r) * DD);
    float4 xc[4], xp[4];
    float x2 = 0.f, y2 = 0.f, xy = 0.f;
#pragma unroll
    for (int i = 0; i < 4; ++i) {
      xc[i] = xc4[lane + 32 * i];
      xp[i] = xp4[lane + 32 * i];
      x2 += d4(xp[i], xp[i]);
      y2 += d4(xc[i], xc[i]);
      xy -= d4(xp[i], xc[i]);             // x = -x_prev
    }
    x2 = wred(x2); y2 = wred(y2); xy = wred(xy);
    // sub = mobius_add(-xp, xc)
    const float den = fmaxf(1.0f + 2.0f * xy + x2 * y2, FEPS);
    const float ca  = (1.0f + 2.0f * xy + y2) / den;   // times (-xp)
    const float cb  = (1.0f - x2) / den;               // times xc
    float4 sb[4];
    float s2 = 0.f;
#pragma unroll
    for (int i = 0; i < 4; ++i) {
      sb[i] = axpby4(-ca, xp[i], cb, xc[i]);
      s2 += d4(sb[i], sb[i]);
    }
    s2 = wred(s2);
    const float sn  = sqrtf(fmaxf(s2, FEPS));
    const float lam = 2.0f / fmaxf(1.0f - x2, FEPS);
    const float sbt = (2.0f / lam) * atanhc_(sn) / sn;   // bt = sbt * sb
    const float un  = sqrtf(fmaxf(sbt * sbt * s2, FEPS));
    const float th  = tanhf(un);
    const float su  = th / un * sbt;                     // u = su * sb
#pragma unroll
    for (int i = 0; i < 4; ++i)
      *(float4*)&uS[r * USTR + 4 * (lane + 32 * i)] = ax4(su, sb[i]);
    if (lane == 0) cxS[r] = th * th;                     // cx2 = |u|^2
  }

  // ===== Phase 2/3: WMMA matmul over 8 z panels + sinh epilogue ============
  const int m0   = (wave >> 2) * 16;       // row group: 0 or 16
  const int nl0  = (wave & 3) * 16;        // col group inside panel
  const int half = lane >> 4;
  const int l16  = lane & 15;

  for (int p = 0; p < DD / PANEL; ++p) {
    const int n0 = p * PANEL;
    __syncthreads();                       // everyone done with previous zS
    if (tid == 0) {
      // Tensor Data Mover: DMA z[0:512][n0:n0+64] (row-major, stride 512) -> zS
      unsigned long long ga = (unsigned long long)(uintptr_t)(const void*)(z + n0);
      v4u g0;
      g0.x = 1u;                                           // count=1, user descriptor
      g0.y = (unsigned)ZS_OFF;                             // lds_addr (bytes)
      g0.z = (unsigned)(ga & 0xFFFFFFFFull);               // global_addr[31:0]
      g0.w = (unsigned)((ga >> 32) & 0x1FFFFFFull) | (2u << 30);  // addr[56:32] | type=2
      v8i g1;
      g1[0] = (int)(2u << 16);                             // data_size = 4B
      g1[1] = (int)(((unsigned)DD & 0xFFFFu) << 16);       // tensor_dim0 lo16 @ [63:48]
      g1[2] = (int)(((unsigned)DD & 0xFFFFu) << 16);       // dim0 hi16=0; tensor_dim1 lo16 @ [95:80]
      g1[3] = (int)(((unsigned)PANEL) << 16);              // dim1 hi16=0; tile_dim0=64 @ [127:112]
      g1[4] = (int)DD;                                     // tile_dim1=512; tile_dim2=0
      g1[5] = (int)DD;                                     // tensor_dim0_stride = 512
      g1[6] = 0; g1[7] = 0;                                // stride0 hi, dim1_stride = 0
      v4i gz4 = {0, 0, 0, 0};
      v8i gz8 = {0, 0, 0, 0, 0, 0, 0, 0};
      __builtin_amdgcn_tensor_load_to_lds(g0, g1, gz4, gz4, gz8, 0);
      __builtin_amdgcn_s_wait_tensorcnt(0);
    }
    __syncthreads();                       // zS panel ready for all waves

    // 16x16 output tile per wave, K-loop of f32 WMMA (K step = 4)
    v8f acc = {0.f, 0.f, 0.f, 0.f, 0.f, 0.f, 0.f, 0.f};
    const float* uRow = &uS[(m0 + l16) * USTR + 2 * half];   // A: row m0+l16, k+2*half
    const float* zCol = &zS[(2 * half) * PANEL + nl0 + l16]; // B: col nl0+l16, k+2*half
#pragma unroll 8
    for (int k = 0; k < DD; k += 4) {
      v2f A = *(const v2f*)(uRow + k);
      v2f Bv;
      Bv.x = zCol[k * PANEL];
      Bv.y = zCol[k * PANEL + PANEL];
      acc = __builtin_amdgcn_wmma_f32_16x16x4_f32(
          /*neg_a=*/false, A, /*neg_b=*/false, Bv,
          /*c_mod=*/(short)0, acc, /*reuse_a=*/false, /*reuse_b=*/false);
    }

    // panel epilogue: inner -> w = sinh(2*zn*asinh(...)); zn-normalization folded here
    const int col = n0 + nl0 + l16;
    const float znc = zn[col], rz = rzn[col], ch = c2r[col], sh = s2r[col];
#pragma unroll
    for (int i = 0; i < 8; ++i) {
      const int r = m0 + i + 8 * half;     // C/D layout: VGPR i, lanes 16-31 -> M+8
      const float cx2 = cxS[r];
      const float inner = acc[i] * rz;
      const float vv = 2.0f * znc *
          asinhf((2.0f * inner * ch - (1.0f + cx2) * sh) / fmaxf(1.0f - cx2, FEPS));
      wS[r * USTR + col] = sinhf(vv);
    }
  }
  __syncthreads();

  // ===== Phase 4: fc denom + logmap0 + expmap + residual update ============
  const float a    = sigmoidf_(alpha_p[0]);
  const float step = sigmoidf_(step_p[0]);
  for (int rr = 0; rr < 4; ++rr) {
    const int r = wave * 4 + rr;
    const float4* xc4 = (const float4*)(xcur + (size_t)(row0 + r) * DD);
    const float4* w4  = (const float4*)(wS + r * USTR);
    float4 xc[4], w[4];
    float s = 0.f, x2c = 0.f;
#pragma unroll
    for (int i = 0; i < 4; ++i) {
      w[i]  = w4[lane + 32 * i];
      xc[i] = xc4[lane + 32 * i];
      s   += d4(w[i], w[i]);
      x2c += d4(xc[i], xc[i]);
    }
    s = wred(s); x2c = wred(x2c);
    // poincare_fc tail: y = w / (1 + sqrt(1 + |w|^2));  logmap0: vel = atanh(|y|)*y/|y|
    const float wd  = 1.0f / (1.0f + sqrtf(1.0f + s));
    const float yn  = sqrtf(fmaxf(s * wd * wd, FEPS));
    const float art = atanhc_(yn);
    const float fu  = step * art / yn * wd;                 // u = fu * w  (u = step*vel)
    const float un  = sqrtf(fmaxf(step * step * art * art, FEPS));
    float xyu = 0.f;
#pragma unroll
    for (int i = 0; i < 4; ++i) xyu += d4(xc[i], w[i]);
    xyu = wred(xyu) * fu;                                   // <xc, u>
    // expmap(xc, u): t = tanh(lam*un/2)*u/un; mobius_add(xc, t)
    const float lam = 2.0f / fmaxf(1.0f - x2c, FEPS);
    const float tht = tanhf(0.5f * lam * un);
    const float ft  = tht / un * fu;                        // t = ft * w
    const float y2t = tht * tht;                            // |t|^2
    const float xyt = tht / un * xyu;
    const float d1  = fmaxf(1.0f + 2.0f * xyt + x2c * y2t, FEPS);
    const float e1  = (1.0f + 2.0f * xyt + y2t) / d1;       // * xc
    const float e2  = (1.0f - x2c) / d1 * ft;               // * w
    float4 xu[4];
    float su2 = 0.f, cxu = 0.f;
#pragma unroll
    for (int i = 0; i < 4; ++i) {
      xu[i] = axpby4(e1, xc[i], e2, w[i]);
      su2 += d4(xu[i], xu[i]);
      cxu += d4(xc[i], xu[i]);
    }
    su2 = wred(su2); cxu = wred(cxu);
    // projx on x_update and on x_current
    const float nu = sqrtf(fmaxf(su2, FEPS));
    const float pu = (nu > PMAXN) ? (PMAXN / nu) : 1.0f;
    const float nup = fminf(nu, PMAXN);
    const float nc = sqrtf(fmaxf(x2c, FEPS));
    const float pc = (nc > PMAXN) ? (PMAXN / nc) : 1.0f;
    const float ncp = fminf(nc, PMAXN);
    // mobius scalar muls: m1 = tanh(a*atanh(|xcp|)) * xcp/|xcp|, m2 likewise (1-a)
    const float xnc = sqrtf(fmaxf(ncp * ncp, FEPS));
    const float T1  = tanhf(a * atanhc_(xnc));
    const float f1  = T1 * pc / xnc;                        // m1 = f1 * xc
    const float xnu = sqrtf(fmaxf(nup * nup, FEPS));
    const float T2  = tanhf((1.0f - a) * atanhc_(xnu));
    const float f2  = T2 * pu / xnu;                        // m2 = f2 * xu
    // mobius_add(m1, m2)
    const float x2m = T1 * T1, y2m = T2 * T2;
    const float xym = f1 * f2 * cxu;
    const float d2  = fmaxf(1.0f + 2.0f * xym + x2m * y2m, FEPS);
    const float g1c = (1.0f + 2.0f * xym + y2m) * f1 / d2;  // * xc
    const float g2c = (1.0f - x2m) * f2 / d2;               // * xu
    float4 xr[4];
    float sr2 = 0.f;
#pragma unroll
    for (int i = 0; i < 4; ++i) {
      xr[i] = axpby4(g1c, xc[i], g2c, xu[i]);
      sr2 += d4(xr[i], xr[i]);
    }
    sr2 = wred(sr2);
    const float nr = sqrtf(fmaxf(sr2, FEPS));
    const float pr = (nr > PMAXN) ? (PMAXN / nr) : 1.0f;
    // outputs: x_next, then xc = projx(x_current)
    float4* o1 = (float4*)(out + (size_t)(row0 + r) * DD);
    float4* o2 = (float4*)(out + (size_t)BN * DD + (size_t)(row0 + r) * DD);
#pragma unroll
    for (int i = 0; i < 4; ++i) {
      o1[lane + 32 * i] = ax4(pr, xr[i]);
      o2[lane + 32 * i] = ax4(pc, xc[i]);
    }
  }
}

// ---------------------------- launcher -------------------------------------
extern "C" void kernel_launch(void* const* d_in, const int* in_sizes, int n_in,
                              void* d_out, int out_size, void* d_ws, size_t ws_size,
                              hipStream_t stream) {
  const float* xcur  = (const float*)d_in[0];
  const float* xprev = (const float*)d_in[1];
  const float* z     = (const float*)d_in[2];
  const float* bias  = (const float*)d_in[3];
  const float* alpha = (const float*)d_in[4];
  const float* stepz = (const float*)d_in[5];
  float* out = (float*)d_out;

  float* zn  = (float*)d_ws;        // 512
  float* rzn = zn + DD;             // 512
  float* c2r = zn + 2 * DD;         // 512
  float* s2r = zn + 3 * DD;         // 512  (8 KB total workspace)

  hpd_prep<<<(DD + 255) / 256, 256, 0, stream>>>(z, bias, zn, rzn, c2r, s2r);

  (void)hipFuncSetAttribute((const void*)hpd_main,
                            hipFuncAttributeMaxDynamicSharedMemorySize, SMEM_B);
  hpd_main<<<BN / ROWS, 256, SMEM_B, stream>>>(xcur, xprev, z, zn, rzn, c2r, s2r,
                                               alpha, stepz, out);
}